// project_C_grasp_12610023981115
// MI455X (gfx1250) — compile-verified
//
#include <hip/hip_runtime.h>

// ---------------------------------------------------------------------------
// MI455X (gfx1250) bandwidth-bound gather/scatter constraint projection.
// Outputs: d_out = [ V_predict_new (N3 floats) | L_new (K floats) ].
// Roofline: ~330 MB total traffic, ~50 MFLOP -> pure HBM-bound (~14 us at
// 23.3 TB/s). WMMA is inapplicable (no matmul structure); the CDNA5 paths
// that matter are data movement:
//   Kernel 1: stream-copy V_predict -> out via async global->LDS->global DMA
//             (ASYNCcnt path, b128 per lane), NT fallback otherwise.
//   Kernel 2: per-constraint projection; 2 constraints/thread with
//             global_prefetch_b8 software prefetch of the second gather.
// ---------------------------------------------------------------------------

typedef __attribute__((ext_vector_type(4))) float f4;
// Exact element type the async-LDS builtins expect (int4).
typedef __attribute__((__vector_size__(4 * sizeof(int)))) int v4i;

#define AS1 __attribute__((address_space(1)))
#define AS3 __attribute__((address_space(3)))

#if defined(__gfx1250__) && \
    __has_builtin(__builtin_amdgcn_global_load_async_to_lds_b128) && \
    __has_builtin(__builtin_amdgcn_global_store_async_from_lds_b128)
#define USE_ASYNC_LDS_COPY 1
#else
#define USE_ASYNC_LDS_COPY 0
#endif

__device__ __forceinline__ void wait_asynccnt0() {
#if defined(__gfx1250__)
#if __has_builtin(__builtin_amdgcn_s_wait_asynccnt)
  __builtin_amdgcn_s_wait_asynccnt(0);
#else
  asm volatile("s_wait_asynccnt 0" ::: "memory");
#endif
#endif
}

// ---------------------------------------------------------------------------
// Kernel 1: copy V_predict (N3 floats, N3 % 4 == 0 for the main body) to out.
// ---------------------------------------------------------------------------
__global__ __launch_bounds__(256) void copy_v_kernel(
    const f4* __restrict__ src4, f4* __restrict__ dst4, int n4,
    const float* __restrict__ srcf, float* __restrict__ dstf, int ntail) {
  int i = blockIdx.x * 256 + threadIdx.x;

#if USE_ASYNC_LDS_COPY
  __shared__ f4 stage[256];
  if (i < n4) {
    // Generic->AS pointers via integer reinterpret (LDS generic addr low 32
    // bits == LDS byte offset; global generic addr == AS1 addr).
    AS1 v4i* gsrc = (AS1 v4i*)(unsigned long long)(const void*)(src4 + i);
    AS1 v4i* gdst = (AS1 v4i*)(unsigned long long)(void*)(dst4 + i);
    AS3 v4i* lds =
        (AS3 v4i*)(unsigned)(unsigned long long)(void*)(stage + threadIdx.x);
    // HBM -> LDS (b128/lane), drain ASYNCcnt, LDS -> HBM (b128/lane).
    __builtin_amdgcn_global_load_async_to_lds_b128(gsrc, lds, 0, 0);
    wait_asynccnt0();
    __builtin_amdgcn_global_store_async_from_lds_b128(gdst, lds, 0, 0);
    // S_ENDPGM implicitly waits for all counters.
  }
#else
  if (i < n4) {
    // Regular-temporal load (warm L2 for the gather kernel), NT store
    // (output is never re-read on device; keep V_predict L2-resident).
    f4 v = src4[i];
    __builtin_nontemporal_store(v, dst4 + i);
  }
#endif

  // Scalar tail (N3 % 4), done by one thread; N3 = 24M -> ntail == 0.
  if (i == 0) {
    for (int t = 0; t < ntail; ++t) {
      size_t e = (size_t)n4 * 4 + (size_t)t;
      dstf[e] = srcf[e];
    }
  }
}

// ---------------------------------------------------------------------------
// Kernel 2: constraint projection (one constraint per call of this helper).
// ---------------------------------------------------------------------------
__device__ __forceinline__ void project_one(
    int k, int v,
    const float* __restrict__ V, const float* __restrict__ L,
    const float* __restrict__ Vw, const float* __restrict__ GP,
    const float* __restrict__ GD, float* __restrict__ outV,
    float* __restrict__ outL) {
  size_t v3 = 3u * (size_t)v;
  size_t k3 = 3u * (size_t)k;

  float px = V[v3 + 0];
  float py = V[v3 + 1];
  float pz = V[v3 + 2];

  float nx = px - GP[k3 + 0];
  float ny = py - GP[k3 + 1];
  float nz = pz - GP[k3 + 2];

  float D  = sqrtf(nx * nx + ny * ny + nz * nz);
  float C  = D - GD[k];
  float lk = L[k];
  float Sw = Vw[v];
  // pinned vertex (w == 0) -> S = inf -> L_delta = +-0 (matches reference)
  float S  = (Sw == 0.0f) ? __builtin_inff() : Sw;
  float Ld = (-C - lk) / (S + 1.0f);   // A = 1.0

  outL[k] = lk + Ld;

  float s = Sw * Ld / D;               // Sw * L_delta * (N / D)
  outV[v3 + 0] = px + s * nx;
  outV[v3 + 1] = py + s * ny;
  outV[v3 + 2] = pz + s * nz;
}

__global__ __launch_bounds__(256) void grasp_project_kernel(
    const float* __restrict__ V, const float* __restrict__ L,
    const float* __restrict__ Vw, const int* __restrict__ Cidx,
    const float* __restrict__ GP, const float* __restrict__ GD,
    float* __restrict__ outV, float* __restrict__ outL, int K, int half) {
  int t = blockIdx.x * 256 + threadIdx.x;
  if (t >= half) return;

  int k0 = t;
  int k1 = t + half;

  int i0 = Cidx[k0];
  int i1 = (k1 < K) ? Cidx[k1] : i0;

  // Software-prefetch the second constraint's random gather targets into the
  // cache hierarchy (global_prefetch_b8) while the first one computes.
  __builtin_prefetch(&V[3u * (size_t)i1], 0, 3);
  __builtin_prefetch(&Vw[i1], 0, 3);

  project_one(k0, i0, V, L, Vw, GP, GD, outV, outL);
  if (k1 < K) {
    project_one(k1, i1, V, L, Vw, GP, GD, outV, outL);
  }
}

// ---------------------------------------------------------------------------
// Launch: copy first, then project (stream order makes the non-atomic
// overwrite of constrained vertices correct; indices are unique).
// ---------------------------------------------------------------------------
extern "C" void kernel_launch(void* const* d_in, const int* in_sizes, int n_in,
                              void* d_out, int out_size, void* d_ws,
                              size_t ws_size, hipStream_t stream) {
  (void)n_in; (void)out_size; (void)d_ws; (void)ws_size;

  const float* V  = (const float*)d_in[0];  // V_predict   [N,3] flat
  const float* L  = (const float*)d_in[1];  // L           [K]
  const float* Vw = (const float*)d_in[2];  // V_w         [N]
  const int*   Ci = (const int*)d_in[3];    // C_grasp     [K] (int32 per harness)
  const float* GP = (const float*)d_in[4];  // grasp_point [K,3] flat
  const float* GD = (const float*)d_in[5];  // C_grasp_d   [K]

  const int N3 = in_sizes[0];               // 24,000,000
  const int K  = in_sizes[1];               //  2,000,000

  float* outV = (float*)d_out;
  float* outL = outV + (size_t)N3;

  const int n4    = N3 / 4;
  const int ntail = N3 - 4 * n4;
  const int cblocks = (n4 + 255) / 256;
  copy_v_kernel<<<cblocks, 256, 0, stream>>>(
      (const f4*)V, (f4*)outV, n4, V, outV, ntail);

  const int half = (K + 1) / 2;
  const int gblocks = (half + 255) / 256;
  grasp_project_kernel<<<gblocks, 256, 0, stream>>>(
      V, L, Vw, Ci, GP, GD, outV, outL, K, half);
}